// GAT_27951647162850
// MI455X (gfx1250) — compile-verified
//
#include <hip/hip_runtime.h>
#include <math.h>

typedef float v2f __attribute__((ext_vector_type(2)));
typedef float v8f __attribute__((ext_vector_type(8)));

#define N_NODES 50000
#define E_EDGES 800000
#define EP      (E_EDGES + N_NODES)   // edges + self loops
#define IN_C    256
#define HID     32
#define HEADS   8
#define H1      (HEADS * HID)         // 256
#define OUT_C   16
#define NEG_SLOPE 0.2f

__device__ __forceinline__ float lrelu(float v) { return v >= 0.f ? v : NEG_SLOPE * v; }

// ---------------------------------------------------------------------------
// C[M,Nc] = A[M,K] @ B[K,Nc], row-major fp32, via V_WMMA_F32_16X16X4_F32.
// One wave per 16x16 output tile. Fragment layouts per CDNA5 ISA 7.12.2:
//   A 16x4 : lane l (grp=lane>>4): a = {A[mt*16+l][k+2g], A[..][k+2g+1]}
//   B 4x16 : lane l: b = {B[k+2g][nt*16+l], B[k+2g+1][nt*16+l]}
//   C 16x16: vgpr i: row = mt*16 + i + 8*grp, col = nt*16 + l
// ---------------------------------------------------------------------------
__global__ void gemm_wmma_f32(const float* __restrict__ A, const float* __restrict__ B,
                              float* __restrict__ C, int M, int K, int Nc) {
  int wave = (blockIdx.x * blockDim.x + threadIdx.x) >> 5;
  int lane = threadIdx.x & 31;
  int tilesN = Nc >> 4;
  int tilesM = M >> 4;
  if (wave >= tilesM * tilesN) return;
  int mt = wave / tilesN, nt = wave % tilesN;
  int grp = lane >> 4;
  int l   = lane & 15;
  int col = (nt << 4) + l;
  const float* arow = A + (size_t)((mt << 4) + l) * K;
  v8f acc = {0.f, 0.f, 0.f, 0.f, 0.f, 0.f, 0.f, 0.f};
  for (int k = 0; k < K; k += 4) {
    int kk = k + 2 * grp;
    v2f a = *(const v2f*)(arow + kk);
    v2f b;
    b.x = B[(size_t)kk * Nc + col];
    b.y = B[(size_t)(kk + 1) * Nc + col];
    acc = __builtin_amdgcn_wmma_f32_16x16x4_f32(false, a, false, b, (short)0, acc,
                                                false, false);
  }
  for (int i = 0; i < 8; ++i)
    C[(size_t)((mt << 4) + i + 8 * grp) * Nc + col] = acc[i];
}

// a_src1[n,h] = dot(xl[n,h,:], att_src[h,:]); same for a_dst1. One thread per (n,h).
__global__ void att_scores1(const float* __restrict__ xl, const float* __restrict__ aws,
                            const float* __restrict__ awd, float* __restrict__ as,
                            float* __restrict__ ad) {
  int t = blockIdx.x * blockDim.x + threadIdx.x;
  if (t >= N_NODES * HEADS) return;
  int n = t / HEADS, h = t % HEADS;
  const float* xp = xl + (size_t)n * H1 + h * HID;
  const float* ws = aws + h * HID;
  const float* wd = awd + h * HID;
  float s = 0.f, d = 0.f;
  for (int c = 0; c < HID; ++c) { float v = xp[c]; s += v * ws[c]; d += v * wd[c]; }
  as[t] = s; ad[t] = d;
}

// denom[d,h] += exp(leakyrelu(a_src[s,h] + a_dst[d,h]))  (max-shift cancels exactly)
__global__ void edge_denom1(const int* __restrict__ ei, const float* __restrict__ as,
                            const float* __restrict__ ad, float* __restrict__ den) {
  int t = blockIdx.x * blockDim.x + threadIdx.x;
  if (t >= EP * HEADS) return;
  int e = t >> 3, h = t & 7;
  int s, d;
  if (e < E_EDGES) { s = ei[e]; d = ei[E_EDGES + e]; } else { s = d = e - E_EDGES; }
  float v = as[s * HEADS + h] + ad[d * HEADS + h];
  atomicAdd(&den[d * HEADS + h], expf(lrelu(v)));
}

// One wave per edge; lane owns 8 contiguous channels (head = lane/4).
__global__ void edge_agg1(const int* __restrict__ ei, const float* __restrict__ as,
                          const float* __restrict__ ad, const float* __restrict__ den,
                          const float* __restrict__ xl, float* __restrict__ outh) {
  int gt = blockIdx.x * blockDim.x + threadIdx.x;
  int wave = gt >> 5, lane = gt & 31;
  if (wave >= EP) return;
  int s, d;
  if (wave < E_EDGES) { s = ei[wave]; d = ei[E_EDGES + wave]; }
  else { s = d = wave - E_EDGES; }
  int h = lane >> 2;
  float v = as[s * HEADS + h] + ad[d * HEADS + h];
  float alpha = expf(lrelu(v)) / (den[d * HEADS + h] + 1e-16f);
  const float4* xs = (const float4*)(xl + (size_t)s * H1 + lane * 8);
  float4 x0 = xs[0], x1 = xs[1];
  float* op = outh + (size_t)d * H1 + lane * 8;
  atomicAdd(op + 0, alpha * x0.x); atomicAdd(op + 1, alpha * x0.y);
  atomicAdd(op + 2, alpha * x0.z); atomicAdd(op + 3, alpha * x0.w);
  atomicAdd(op + 4, alpha * x1.x); atomicAdd(op + 5, alpha * x1.y);
  atomicAdd(op + 6, alpha * x1.z); atomicAdd(op + 7, alpha * x1.w);
}

__global__ void bias_relu(float* __restrict__ h, const float* __restrict__ b) {
  int t = blockIdx.x * blockDim.x + threadIdx.x;
  if (t >= N_NODES * H1) return;
  float v = h[t] + b[t % H1];
  h[t] = v > 0.f ? v : 0.f;
}

__global__ void att_scores2(const float* __restrict__ xl, const float* __restrict__ aws,
                            const float* __restrict__ awd, float* __restrict__ as,
                            float* __restrict__ ad) {
  int t = blockIdx.x * blockDim.x + threadIdx.x;
  if (t >= N_NODES) return;
  const float* xp = xl + (size_t)t * OUT_C;
  float s = 0.f, d = 0.f;
  for (int c = 0; c < OUT_C; ++c) { float v = xp[c]; s += v * aws[c]; d += v * awd[c]; }
  as[t] = s; ad[t] = d;
}

__global__ void edge_denom2(const int* __restrict__ ei, const float* __restrict__ as,
                            const float* __restrict__ ad, float* __restrict__ den) {
  int e = blockIdx.x * blockDim.x + threadIdx.x;
  if (e >= EP) return;
  int s, d;
  if (e < E_EDGES) { s = ei[e]; d = ei[E_EDGES + e]; } else { s = d = e - E_EDGES; }
  atomicAdd(&den[d], expf(lrelu(as[s] + ad[d])));
}

__global__ void edge_agg2(const int* __restrict__ ei, const float* __restrict__ as,
                          const float* __restrict__ ad, const float* __restrict__ den,
                          const float* __restrict__ xl, float* __restrict__ o2) {
  int t = blockIdx.x * blockDim.x + threadIdx.x;
  if (t >= EP * OUT_C) return;
  int e = t >> 4, c = t & 15;
  int s, d;
  if (e < E_EDGES) { s = ei[e]; d = ei[E_EDGES + e]; } else { s = d = e - E_EDGES; }
  float alpha = expf(lrelu(as[s] + ad[d])) / (den[d] + 1e-16f);
  atomicAdd(&o2[(size_t)d * OUT_C + c], alpha * xl[(size_t)s * OUT_C + c]);
}

__global__ void bias_log_softmax16(const float* __restrict__ o2,
                                   const float* __restrict__ b2,
                                   float* __restrict__ out) {
  int n = blockIdx.x * blockDim.x + threadIdx.x;
  if (n >= N_NODES) return;
  float v[OUT_C];
  float m = -1e30f;
  for (int c = 0; c < OUT_C; ++c) {
    v[c] = o2[(size_t)n * OUT_C + c] + b2[c];
    m = fmaxf(m, v[c]);
  }
  float sum = 0.f;
  for (int c = 0; c < OUT_C; ++c) sum += expf(v[c] - m);
  float lse = m + logf(sum);
  for (int c = 0; c < OUT_C; ++c) out[(size_t)n * OUT_C + c] = v[c] - lse;
}

extern "C" void kernel_launch(void* const* d_in, const int* in_sizes, int n_in,
                              void* d_out, int out_size, void* d_ws, size_t ws_size,
                              hipStream_t stream) {
  const float* x   = (const float*)d_in[0];
  const int*   ei  = (const int*)d_in[1];
  const float* W1  = (const float*)d_in[2];
  const float* as1 = (const float*)d_in[3];
  const float* ad1 = (const float*)d_in[4];
  const float* b1  = (const float*)d_in[5];
  const float* W2  = (const float*)d_in[6];
  const float* as2 = (const float*)d_in[7];
  const float* ad2 = (const float*)d_in[8];
  const float* b2  = (const float*)d_in[9];
  float* out = (float*)d_out;

  float* ws = (float*)d_ws;
  size_t off = 0;
  float* xl1  = ws + off; off += (size_t)N_NODES * H1;      // x @ W1
  float* h1   = ws + off; off += (size_t)N_NODES * H1;      // layer-1 output / hidden
  float* asc1 = ws + off; off += (size_t)N_NODES * HEADS;
  float* adc1 = ws + off; off += (size_t)N_NODES * HEADS;
  float* den1 = ws + off; off += (size_t)N_NODES * HEADS;
  float* xl2  = ws + off; off += (size_t)N_NODES * OUT_C;   // h @ W2
  float* asc2 = ws + off; off += (size_t)N_NODES;
  float* adc2 = ws + off; off += (size_t)N_NODES;
  float* den2 = ws + off; off += (size_t)N_NODES;
  float* o2   = ws + off; off += (size_t)N_NODES * OUT_C;

  hipMemsetAsync(h1,   0, (size_t)N_NODES * H1 * sizeof(float), stream);
  hipMemsetAsync(den1, 0, (size_t)N_NODES * HEADS * sizeof(float), stream);
  hipMemsetAsync(den2, 0, (size_t)N_NODES * sizeof(float), stream);
  hipMemsetAsync(o2,   0, (size_t)N_NODES * OUT_C * sizeof(float), stream);

  const int BT = 256;
  auto blk = [](long long n) { return (int)((n + 255) / 256); };

  // ---- layer 1 ----
  long long w1waves = (long long)(N_NODES / 16) * (H1 / 16);          // 50000 waves
  gemm_wmma_f32<<<blk(w1waves * 32), BT, 0, stream>>>(x, W1, xl1, N_NODES, IN_C, H1);
  att_scores1<<<blk((long long)N_NODES * HEADS), BT, 0, stream>>>(xl1, as1, ad1, asc1, adc1);
  edge_denom1<<<blk((long long)EP * HEADS), BT, 0, stream>>>(ei, asc1, adc1, den1);
  edge_agg1<<<blk((long long)EP * 32), BT, 0, stream>>>(ei, asc1, adc1, den1, xl1, h1);
  bias_relu<<<blk((long long)N_NODES * H1), BT, 0, stream>>>(h1, b1);

  // ---- layer 2 ----
  long long w2waves = (long long)(N_NODES / 16) * (OUT_C / 16);       // 3125 waves
  gemm_wmma_f32<<<blk(w2waves * 32), BT, 0, stream>>>(h1, W2, xl2, N_NODES, H1, OUT_C);
  att_scores2<<<blk((long long)N_NODES), BT, 0, stream>>>(xl2, as2, ad2, asc2, adc2);
  edge_denom2<<<blk((long long)EP), BT, 0, stream>>>(ei, asc2, adc2, den2);
  edge_agg2<<<blk((long long)EP * OUT_C), BT, 0, stream>>>(ei, asc2, adc2, den2, xl2, o2);
  bias_log_softmax16<<<blk((long long)N_NODES), BT, 0, stream>>>(o2, b2, out);
}